// GCN_37632503448199
// MI455X (gfx1250) — compile-verified
//
#include <hip/hip_runtime.h>
#include <hip/hip_bf16.h>

typedef __attribute__((ext_vector_type(2))) float v2f;
typedef __attribute__((ext_vector_type(8))) float v8f;

#define TPB 256

__device__ __forceinline__ void atomAddF(float* p, float v) {
  // fp32 hardware atomic (global_atomic_add_f32), relaxed, device scope
  __hip_atomic_fetch_add(p, v, __ATOMIC_RELAXED, __HIP_MEMORY_SCOPE_AGENT);
}

// ---------------- degree / normalization ----------------

__global__ void k_fill(float* __restrict__ p, float v, int n) {
  int i = blockIdx.x * blockDim.x + threadIdx.x;
  if (i < n) p[i] = v;
}

__global__ void k_degree(const long long* __restrict__ dst, float* __restrict__ deg, int e) {
  int i = blockIdx.x * blockDim.x + threadIdx.x;
  if (i < e) atomAddF(&deg[(int)dst[i]], 1.0f);
}

__global__ void k_rsqrt_inplace(float* __restrict__ p, int n) {
  int i = blockIdx.x * blockDim.x + threadIdx.x;
  if (i < n) p[i] = rsqrtf(p[i]);
}

// ---------------- WMMA GEMM: C[M x 16] = A[M x K] * B[K x 16] ----------------
// One wave per 16x16 output tile using V_WMMA_F32_16X16X4_F32 (exact fp32).
// A 16x4 layout: lanes 0-15: M=lane, VGPR0=K0,VGPR1=K1; lanes 16-31: K2/K3.
// B 4x16 layout: lanes 0-15: N=lane, K0/K1; lanes 16-31: K2/K3.
// C/D: VGPR j = row (j + 8*(lane>=16)), col = lane&15.
// B is pre-swizzled into LDS once per block (8 waves share it): BL[k0/4][lane]
// holds the float2 {B[k0+khalf][col], B[k0+khalf+1][col]} that lane needs.

__global__ void k_gemm_wmma(const float* __restrict__ A,
                            const float* __restrict__ B,
                            float* __restrict__ C,
                            int M, int K) {
  __shared__ float2 BL[4096];          // up to K=512: (512/4)*32 = 4096 float2 = 32 KB

  const int tid  = threadIdx.x;
  const int lane = tid & 31;

  // ---- stage swizzled B into LDS (whole block cooperates) ----
  const int nEnt = (K >> 2) * 32;
  for (int idx = tid; idx < nEnt; idx += TPB) {
    int step  = idx >> 5;              // k-block index (K step of 4)
    int l     = idx & 31;
    int col   = l & 15;
    int kh    = (l >> 4) << 1;
    int k     = (step << 2) + kh;
    float2 w;
    w.x = B[k * 16 + col];
    w.y = B[(k + 1) * 16 + col];
    BL[idx] = w;
  }
  __syncthreads();

  const int wave = (int)((blockIdx.x * blockDim.x + tid) >> 5);
  const int m0   = wave * 16;
  if (m0 >= M) return;                 // per-wave uniform: EXEC all-ones below

  const int col   = lane & 15;
  const int khalf = (lane >> 4) << 1;

  int row = m0 + col;
  if (row >= M) row = M - 1;           // clamp (keep EXEC full for WMMA)
  const float* arow = A + (size_t)row * (size_t)K;

  v8f acc = {0.f, 0.f, 0.f, 0.f, 0.f, 0.f, 0.f, 0.f};

  for (int k0 = 0; k0 < K; k0 += 4) {
    v2f a, b;
    a.x = arow[k0 + khalf + 0];
    a.y = arow[k0 + khalf + 1];
    float2 w = BL[(k0 >> 2) * 32 + lane];
    b.x = w.x;
    b.y = w.y;
    acc = __builtin_amdgcn_wmma_f32_16x16x4_f32(
        /*neg_a=*/false, a, /*neg_b=*/false, b,
        /*c_mod=*/(short)0, acc, /*reuse_a=*/false, /*reuse_b=*/false);
  }

  const int rbase = m0 + ((lane >> 4) << 3);
#pragma unroll
  for (int j = 0; j < 8; ++j) {
    int r = rbase + j;
    if (r < M) C[(size_t)r * 16 + col] = acc[j];
  }
}

// ---------------- GCN aggregation (16 channels, float4-vectorized) ----------------

// agg[i, q..q+3] = h[i, q..q+3] * dis[i]^2   (self-loop term; also inits agg)
__global__ void k_self16(const float* __restrict__ h, const float* __restrict__ dis,
                         float* __restrict__ agg, int n) {
  int t = blockIdx.x * blockDim.x + threadIdx.x;
  if (t >= n * 4) return;
  int i = t >> 2;
  float d = dis[i];
  float dd = d * d;
  float4 hv = ((const float4*)h)[t];
  float4 o;
  o.x = hv.x * dd; o.y = hv.y * dd; o.z = hv.z * dd; o.w = hv.w * dd;
  ((float4*)agg)[t] = o;
}

// agg[dst, q..q+3] += h[src, q..q+3] * dis[src]*dis[dst]   (4 threads per edge)
__global__ void k_edges16(const long long* __restrict__ src, const long long* __restrict__ dst,
                          const float* __restrict__ h, const float* __restrict__ dis,
                          float* __restrict__ agg, int e) {
  int t = blockIdx.x * blockDim.x + threadIdx.x;
  int ei = t >> 2;
  if (ei >= e) return;
  int q = (t & 3) << 2;
  int s = (int)src[ei];
  int d = (int)dst[ei];
  float norm = dis[s] * dis[d];
  float4 hv = *(const float4*)(h + (size_t)s * 16 + q);
  float* ap = agg + (size_t)d * 16 + q;
  atomAddF(ap + 0, hv.x * norm);
  atomAddF(ap + 1, hv.y * norm);
  atomAddF(ap + 2, hv.z * norm);
  atomAddF(ap + 3, hv.w * norm);
}

// out[i, q..q+3] = relu(agg[i, q..q+3] + b[q..q+3])   (in-place safe)
__global__ void k_bias_relu16(const float* __restrict__ agg, const float* __restrict__ b,
                              float* __restrict__ out, int n) {
  int t = blockIdx.x * blockDim.x + threadIdx.x;
  if (t >= n * 4) return;
  int q = (t & 3) << 2;
  float4 v = ((const float4*)agg)[t];
  float4 bb = *(const float4*)(b + q);
  v.x = fmaxf(v.x + bb.x, 0.f);
  v.y = fmaxf(v.y + bb.y, 0.f);
  v.z = fmaxf(v.z + bb.z, 0.f);
  v.w = fmaxf(v.w + bb.w, 0.f);
  ((float4*)out)[t] = v;
}

// ---------------- layer-3 (16 -> 1) ----------------

__global__ void k_dot16(const float* __restrict__ x, const float* __restrict__ w,
                        float* __restrict__ y, int n) {
  int i = blockIdx.x * blockDim.x + threadIdx.x;
  if (i >= n) return;
  const float4* r = (const float4*)(x + (size_t)i * 16);
  const float4* wv = (const float4*)w;
  float s = 0.f;
#pragma unroll
  for (int k = 0; k < 4; ++k) {
    float4 a = r[k], b = wv[k];
    s += a.x * b.x + a.y * b.y + a.z * b.z + a.w * b.w;
  }
  y[i] = s;
}

// agg3[i] = h3[i] * dis[i]^2
__global__ void k_self1(const float* __restrict__ h, const float* __restrict__ dis,
                        float* __restrict__ agg, int n) {
  int i = blockIdx.x * blockDim.x + threadIdx.x;
  if (i >= n) return;
  float d = dis[i];
  agg[i] = h[i] * d * d;
}

__global__ void k_edges1(const long long* __restrict__ src, const long long* __restrict__ dst,
                         const float* __restrict__ h, const float* __restrict__ dis,
                         float* __restrict__ agg, int e) {
  int i = blockIdx.x * blockDim.x + threadIdx.x;
  if (i >= e) return;
  int s = (int)src[i];
  int d = (int)dst[i];
  atomAddF(&agg[d], h[s] * dis[s] * dis[d]);
}

__global__ void k_final_bias(const float* __restrict__ agg, const float* __restrict__ b,
                             float* __restrict__ out, int n) {
  int i = blockIdx.x * blockDim.x + threadIdx.x;
  if (i < n) out[i] = agg[i] + b[0];
}

// ---------------- host launcher ----------------

static inline int gridFor(long long work) { return (int)((work + TPB - 1) / TPB); }

extern "C" void kernel_launch(void* const* d_in, const int* in_sizes, int n_in,
                              void* d_out, int out_size, void* d_ws, size_t ws_size,
                              hipStream_t stream) {
  const float*     x    = (const float*)d_in[0];
  const long long* eidx = (const long long*)d_in[1];
  const float*     W1   = (const float*)d_in[2];
  const float*     b1   = (const float*)d_in[3];
  const float*     W2   = (const float*)d_in[4];
  const float*     b2   = (const float*)d_in[5];
  const float*     W3   = (const float*)d_in[6];
  const float*     b3   = (const float*)d_in[7];

  const int H = in_sizes[3];             // 16
  const int K = in_sizes[2] / H;         // 512
  const int N = in_sizes[0] / K;         // 100000
  const int E = in_sizes[1] / 2;         // 3200000
  (void)H; (void)n_in; (void)ws_size; (void)out_size;

  const long long* srcI = eidx;          // edge_index[0]
  const long long* dstI = eidx + E;      // edge_index[1]

  // workspace: dis[N] | Hbuf[N*16] | Gbuf[N*16] | A3[N]  (~13.6 MB)
  float* ws   = (float*)d_ws;
  float* dis  = ws;
  float* Hbuf = dis + N;
  float* Gbuf = Hbuf + (size_t)N * 16;
  float* A3   = Gbuf + (size_t)N * 16;

  // --- normalization: deg = 1 + count(dst); dis = rsqrt(deg) ---
  k_fill<<<gridFor(N), TPB, 0, stream>>>(dis, 1.0f, N);
  k_degree<<<gridFor(E), TPB, 0, stream>>>(dstI, dis, E);
  k_rsqrt_inplace<<<gridFor(N), TPB, 0, stream>>>(dis, N);

  const int tiles      = (N + 15) / 16;
  const int gemmBlocks = gridFor((long long)tiles * 32);

  // --- layer 1: h1 = X @ W1 ; aggregate ; relu(. + b1) -> Gbuf ---
  k_gemm_wmma<<<gemmBlocks, TPB, 0, stream>>>(x, W1, Hbuf, N, K);
  k_self16<<<gridFor((long long)N * 4), TPB, 0, stream>>>(Hbuf, dis, Gbuf, N);
  k_edges16<<<gridFor((long long)E * 4), TPB, 0, stream>>>(srcI, dstI, Hbuf, dis, Gbuf, E);
  k_bias_relu16<<<gridFor((long long)N * 4), TPB, 0, stream>>>(Gbuf, b1, Gbuf, N);

  // --- layer 2: h2 = x2 @ W2 ; aggregate ; relu(. + b2) -> Gbuf ---
  k_gemm_wmma<<<gemmBlocks, TPB, 0, stream>>>(Gbuf, W2, Hbuf, N, 16);
  k_self16<<<gridFor((long long)N * 4), TPB, 0, stream>>>(Hbuf, dis, Gbuf, N);
  k_edges16<<<gridFor((long long)E * 4), TPB, 0, stream>>>(srcI, dstI, Hbuf, dis, Gbuf, E);
  k_bias_relu16<<<gridFor((long long)N * 4), TPB, 0, stream>>>(Gbuf, b2, Gbuf, N);

  // --- layer 3: h3 = x3 @ W3 (16 -> 1) ; aggregate ; + b3 -> out ---
  k_dot16<<<gridFor(N), TPB, 0, stream>>>(Gbuf, W3, Hbuf, N);   // h3 in Hbuf[0..N)
  k_self1<<<gridFor(N), TPB, 0, stream>>>(Hbuf, dis, A3, N);
  k_edges1<<<gridFor(E), TPB, 0, stream>>>(srcI, dstI, Hbuf, dis, A3, E);
  k_final_bias<<<gridFor(N), TPB, 0, stream>>>(A3, b3, (float*)d_out, N);
}